// SelfAttention_71047349010537
// MI455X (gfx1250) — compile-verified
//
#include <hip/hip_runtime.h>
#include <hip/hip_bf16.h>

typedef float v2f __attribute__((ext_vector_type(2)));
typedef float v8f __attribute__((ext_vector_type(8)));

#define L_SEQ 2048
#define D_HEAD 256
#define NH_ACT 4            // only n<4 of w_prime survives in the reference
#define HEAD_ELEMS ((size_t)L_SEQ * D_HEAD)

static __device__ __forceinline__ v8f wmma_f32(v2f a, v2f b, v8f c) {
    return __builtin_amdgcn_wmma_f32_16x16x4_f32(false, a, false, b, (short)0, c,
                                                 false, false);
}

// ---------------------------------------------------------------- pad lengths
__global__ __launch_bounds__(256) void padlen_kernel(const int* __restrict__ mask,
                                                     int* __restrict__ padlen) {
    __shared__ int red[256];
    const int b = blockIdx.x;
    const int t = threadIdx.x;
    int s = 0;
    for (int i = t; i < L_SEQ; i += 256) s += mask[b * L_SEQ + i];
    red[t] = s;
    __syncthreads();
    for (int off = 128; off > 0; off >>= 1) {
        if (t < off) red[t] += red[t + off];
        __syncthreads();
    }
    if (t == 0) padlen[b] = L_SEQ - red[0];
}

// ------------------------------------------------------- q/k/v projection GEMM
// grid = (128 row tiles, 4 heads, 3 mats), block = 512 (16 waves, 1 col tile each)
__global__ __launch_bounds__(512) void qkv_kernel(const float* __restrict__ x,
                                                  const float* __restrict__ Wq,
                                                  const float* __restrict__ Wk,
                                                  const float* __restrict__ Wv,
                                                  float* __restrict__ q,
                                                  float* __restrict__ k,
                                                  float* __restrict__ v) {
    const int wave = threadIdx.x >> 5;
    const int lane = threadIdx.x & 31;
    const int mrow = lane & 15;            // M (rows of x tile) / N (rows of W tile)
    const int khalf = (lane >> 4) * 2;     // K pair selector per WMMA lane layout
    const int row0 = blockIdx.x * 16;
    const int h = blockIdx.y;
    const int which = blockIdx.z;
    const int col0 = wave * 16;

    const float* W = (which == 0) ? Wq : (which == 1) ? Wk : Wv;
    float* dst = (which == 0) ? q : (which == 1) ? k : v;
    W += (size_t)h * D_HEAD * D_HEAD;      // rows h*256 .. h*256+255 of (2048,256)
    dst += (size_t)h * HEAD_ELEMS;

    const float* arow = x + (size_t)(row0 + mrow) * D_HEAD + khalf;  // batch 0
    const float* brow = W + (size_t)(col0 + mrow) * D_HEAD + khalf;

    v8f acc = {0.f, 0.f, 0.f, 0.f, 0.f, 0.f, 0.f, 0.f};
#pragma unroll 8
    for (int k0 = 0; k0 < D_HEAD; k0 += 4) {
        v2f a = *(const v2f*)(arow + k0);
        v2f b = *(const v2f*)(brow + k0);
        acc = wmma_f32(a, b, acc);
    }
    const int rbase = (lane >> 4) * 8;
#pragma unroll
    for (int r = 0; r < 8; ++r) {
        dst[(size_t)(row0 + r + rbase) * D_HEAD + col0 + mrow] = acc[r];
    }
}

// ------------------------------------------------------- flash attention (f32)
// grid = (128 query tiles, 4 heads), block = 256 (8 waves; wave owns 32 O cols)
__global__ __launch_bounds__(256) void attn_kernel(const float* __restrict__ q,
                                                   const float* __restrict__ k,
                                                   const float* __restrict__ v,
                                                   const int* __restrict__ padlen,
                                                   float* __restrict__ o) {
    const int h = blockIdx.y;
    const int row0 = blockIdx.x * 16;
    const int tid = threadIdx.x;
    const int wave = tid >> 5;
    const int lane = tid & 31;
    const int mrow = lane & 15;
    const int khalf = (lane >> 4) * 2;
    const int rbase = (lane >> 4) * 8;
    const int pl = padlen[h];

    __shared__ __align__(16) float sPart[8][16][16];
    __shared__ __align__(16) float sP[16][16];
    __shared__ float sM[16], sL[16], sAlpha[16], sInvL[16];

    const float* qh = q + (size_t)h * HEAD_ELEMS;
    const float* kh = k + (size_t)h * HEAD_ELEMS;
    const float* vh = v + (size_t)h * HEAD_ELEMS;

    if (tid < 16) { sM[tid] = -1e30f; sL[tid] = 0.f; }
    __syncthreads();

    v8f occ0 = {0.f, 0.f, 0.f, 0.f, 0.f, 0.f, 0.f, 0.f};
    v8f occ1 = occ0;
    const int ocol0 = wave * 32;
    const int dbase = wave * 32;           // this wave's K-slice for Q.K^T

    const int jt_end = row0 >> 4;          // causal tile limit (inclusive)
    for (int jt = pl >> 4; jt <= jt_end; ++jt) {
        const int key0 = jt * 16;

        // ---- partial S over d-slice [dbase, dbase+32)
        v8f s = {0.f, 0.f, 0.f, 0.f, 0.f, 0.f, 0.f, 0.f};
        const float* qp = qh + (size_t)(row0 + mrow) * D_HEAD + dbase + khalf;
        const float* kp = kh + (size_t)(key0 + mrow) * D_HEAD + dbase + khalf;
#pragma unroll
        for (int kk = 0; kk < 8; ++kk) {
            v2f a = *(const v2f*)(qp + kk * 4);
            v2f b = *(const v2f*)(kp + kk * 4);
            s = wmma_f32(a, b, s);
        }
#pragma unroll
        for (int r = 0; r < 8; ++r) sPart[wave][r + rbase][mrow] = s[r];
        __syncthreads();

        // ---- reduce 8 partials -> scaled scores
        {
            const int rr = tid >> 4, cc = tid & 15;
            float sum = 0.f;
#pragma unroll
            for (int wv = 0; wv < 8; ++wv) sum += sPart[wv][rr][cc];
            sP[rr][cc] = sum * 0.0625f;    // 1/sqrt(256)
        }
        __syncthreads();

        // ---- online softmax row update (threads 0..15 own one row each)
        if (tid < 16) {
            const int m = row0 + tid;
            float vals[16];
            float mNew = sM[tid];
#pragma unroll
            for (int c = 0; c < 16; ++c) {
                const int key = key0 + c;
                const bool ok = (m >= pl) && (key >= pl) && (key <= m);
                const float val = ok ? sP[tid][c] : -1e30f;
                vals[c] = val;
                mNew = fmaxf(mNew, val);
            }
            const float alpha = expf(sM[tid] - mNew);
            float lsum = sL[tid] * alpha;
#pragma unroll
            for (int c = 0; c < 16; ++c) {
                const float p = expf(vals[c] - mNew);
                sP[tid][c] = p;
                lsum += p;
            }
            sM[tid] = mNew;
            sL[tid] = lsum;
            sAlpha[tid] = alpha;
        }
        __syncthreads();

        // ---- rescale O and accumulate P @ V[key tile][wave's 32 cols]
#pragma unroll
        for (int r = 0; r < 8; ++r) {
            const float al = sAlpha[r + rbase];
            occ0[r] *= al;
            occ1[r] *= al;
        }
#pragma unroll
        for (int kk = 0; kk < 4; ++kk) {
            v2f a = *(const v2f*)&sP[mrow][kk * 4 + khalf];
            const float* vp = vh + (size_t)(key0 + kk * 4 + khalf) * D_HEAD;
            v2f b0, b1;
            b0.x = vp[ocol0 + mrow];
            b0.y = vp[D_HEAD + ocol0 + mrow];
            b1.x = vp[ocol0 + 16 + mrow];
            b1.y = vp[D_HEAD + ocol0 + 16 + mrow];
            occ0 = wmma_f32(a, b0, occ0);
            occ1 = wmma_f32(a, b1, occ1);
        }
        __syncthreads();
    }

    if (tid < 16) {
        const int m = row0 + tid;
        sInvL[tid] = (m >= pl && sL[tid] > 0.f) ? 1.0f / sL[tid] : 0.0f;
    }
    __syncthreads();

    float* oh = o + (size_t)h * HEAD_ELEMS;
#pragma unroll
    for (int r = 0; r < 8; ++r) {
        const float il = sInvL[r + rbase];
        float* op = oh + (size_t)(row0 + r + rbase) * D_HEAD;
        op[ocol0 + mrow] = occ0[r] * il;
        op[ocol0 + 16 + mrow] = occ1[r] * il;
    }
}

// ---------------------------------------------------- bias fill for 3 batches from offset 2*L*D
__global__ __launch_bounds__(256) void bias_fill(const float* __restrict__ bu,
                                                 float* __restrict__ out) {
    const size_t i = (size_t)blockIdx.x * 256 + threadIdx.x;  // 3*2048*256 elems
    out[HEAD_ELEMS * 2 /* = 2*L*D */ + i] = bu[i & (D_HEAD - 1)];
}

// --------------------------------------- output projection for batch 0 (+ bias)
// out0[row, d] = bu[d] + sum_{e<1024} Oflat[row, e] * Wu[d, e]
// grid = 128 row tiles, block = 512 (16 waves, one 16-col tile each)
__global__ __launch_bounds__(512) void outproj_kernel(const float* __restrict__ o,
                                                      const float* __restrict__ Wu,
                                                      const float* __restrict__ bu,
                                                      float* __restrict__ out) {
    const int wave = threadIdx.x >> 5;
    const int lane = threadIdx.x & 31;
    const int mrow = lane & 15;
    const int khalf = (lane >> 4) * 2;
    const int row0 = blockIdx.x * 16;
    const int col0 = wave * 16;

    v8f acc = {0.f, 0.f, 0.f, 0.f, 0.f, 0.f, 0.f, 0.f};
    const float* wrow = Wu + (size_t)(col0 + mrow) * (8 * D_HEAD) + khalf;
#pragma unroll 4
    for (int k0 = 0; k0 < NH_ACT * D_HEAD; k0 += 4) {
        const int e = k0 + khalf;
        const int hh = e >> 8;
        const int j = e & (D_HEAD - 1);
        v2f a = *(const v2f*)(o + (size_t)hh * HEAD_ELEMS +
                              (size_t)(row0 + mrow) * D_HEAD + j);
        v2f b = *(const v2f*)(wrow + k0);
        acc = wmma_f32(a, b, acc);
    }
    const int rbase = (lane >> 4) * 8;
    const float bias = bu[col0 + mrow];
#pragma unroll
    for (int r = 0; r < 8; ++r) {
        out[(size_t)(row0 + r + rbase) * D_HEAD + col0 + mrow] = acc[r] + bias;
    }
}

// -----------------------------------------------------------------------------
extern "C" void kernel_launch(void* const* d_in, const int* in_sizes, int n_in,
                              void* d_out, int out_size, void* d_ws, size_t ws_size,
                              hipStream_t stream) {
    const float* x    = (const float*)d_in[0];
    const int*   mask = (const int*)d_in[1];
    const float* Wq   = (const float*)d_in[2];
    const float* Wk   = (const float*)d_in[3];
    const float* Wv   = (const float*)d_in[4];
    const float* Wu   = (const float*)d_in[5];
    const float* bu   = (const float*)d_in[6];
    float* out = (float*)d_out;

    float* q = (float*)d_ws;
    float* k = q + NH_ACT * HEAD_ELEMS;
    float* v = k + NH_ACT * HEAD_ELEMS;
    float* o = v + NH_ACT * HEAD_ELEMS;
    int* padlen = (int*)(o + NH_ACT * HEAD_ELEMS);

    padlen_kernel<<<4, 256, 0, stream>>>(mask, padlen);
    qkv_kernel<<<dim3(L_SEQ / 16, NH_ACT, 3), 512, 0, stream>>>(x, Wq, Wk, Wv, q, k, v);
    attn_kernel<<<dim3(L_SEQ / 16, NH_ACT), 256, 0, stream>>>(q, k, v, padlen, o);
    bias_fill<<<(3 * L_SEQ * D_HEAD) / 256, 256, 0, stream>>>(bu, out);
    outproj_kernel<<<dim3(L_SEQ / 16, 1), 512, 0, stream>>>(o, Wu, bu, out);
}